// Sample_Concrete_82617990906605
// MI455X (gfx1250) — compile-verified
//
#include <hip/hip_runtime.h>
#include <stdint.h>

// Sample_Concrete forward: B=1024, D=2048, k=4, K_c=4, tau=0.5
// out[b, d, c] = max_k softmax_c( (gumbel[b,d,k,c] + logits[b,d,c]) / tau )

typedef float v4f __attribute__((ext_vector_type(4)));
typedef int   v4i __attribute__((vector_size(16)));   // matches builtin's V4i param

#define TPB      256
#define NGROUPS  (1024 * 2048)       // 2^21 (b,d) groups, 16 gumbel floats each
#define NTILES   (NGROUPS / TPB)     // 8192
#define NBLOCKS  2048                // each block streams 4 tiles (double-buffered)

#if defined(__gfx1250__) && __has_builtin(__builtin_amdgcn_global_load_async_to_lds_b128)
#define USE_ASYNC 1
#else
#define USE_ASYNC 0
#endif

__device__ __forceinline__ void s_wait_async_le4() {
  asm volatile("s_wait_asynccnt 4" ::: "memory");
}
__device__ __forceinline__ void s_wait_async_le0() {
  asm volatile("s_wait_asynccnt 0" ::: "memory");
}

// softmax over 4 lanes of each row, scaled by 1/tau = 2, running max over 4 rows
__device__ __forceinline__ v4f concrete_group(v4f lg, v4f r0, v4f r1, v4f r2, v4f r3) {
  v4f rows[4] = {r0, r1, r2, r3};
  v4f best = (v4f){0.f, 0.f, 0.f, 0.f};
#pragma unroll
  for (int k = 0; k < 4; ++k) {
    v4f n = (rows[k] + lg) * 2.0f;                       // (g + logits) / tau
    float m = fmaxf(fmaxf(n.x, n.y), fmaxf(n.z, n.w));   // stable softmax
    v4f e;
    e.x = __expf(n.x - m);
    e.y = __expf(n.y - m);
    e.z = __expf(n.z - m);
    e.w = __expf(n.w - m);
    float s   = (e.x + e.y) + (e.z + e.w);
    float inv = __builtin_amdgcn_rcpf(s);
    best.x = fmaxf(best.x, e.x * inv);
    best.y = fmaxf(best.y, e.y * inv);
    best.z = fmaxf(best.z, e.z * inv);
    best.w = fmaxf(best.w, e.w * inv);
  }
  return best;
}

__global__ __launch_bounds__(TPB)
void sample_concrete_kernel(const float* __restrict__ logits,
                            const float* __restrict__ gumbel,
                            float* __restrict__ out) {
  const int tid = (int)threadIdx.x;

#if USE_ASYNC
  // Double-buffered LDS staging for the dominant gumbel stream (64B / thread / tile).
  __shared__ __align__(16) float lds[2][TPB * 16];

  auto issue = [&](int tile, int buf) {
    const size_t goff = ((size_t)tile * TPB + (size_t)tid) * 16;  // floats
    auto g = (__attribute__((address_space(1))) v4i*)(uintptr_t)(gumbel + goff);
    auto l = (__attribute__((address_space(3))) v4i*)(uint32_t)(uintptr_t)(&lds[buf][tid * 16]);
#pragma unroll
    for (int j = 0; j < 4; ++j)
      __builtin_amdgcn_global_load_async_to_lds_b128(g + j, l + j, 0, 0);
  };

  int t   = (int)blockIdx.x;   // < NTILES since NBLOCKS < NTILES
  int buf = 0;
  issue(t, buf);
  for (; t < NTILES; t += (int)gridDim.x) {
    const int  tn = t + (int)gridDim.x;
    const bool pf = (tn < NTILES);
    if (pf) issue(tn, buf ^ 1);
    // Async loads complete in order: <=4 outstanding guarantees current tile landed.
    if (pf) s_wait_async_le4();
    else    s_wait_async_le0();

    const int g  = t * TPB + tid;
    v4f       lg = __builtin_nontemporal_load((const v4f*)logits + g);
    const v4f* lrow = (const v4f*)(&lds[buf][tid * 16]);
    v4f best = concrete_group(lg, lrow[0], lrow[1], lrow[2], lrow[3]);
    __builtin_nontemporal_store(best, (v4f*)out + g);
    buf ^= 1;
  }
#else
  // Fallback: direct streaming b128 loads with non-temporal hints.
  for (int g = (int)blockIdx.x * TPB + tid; g < NGROUPS; g += (int)gridDim.x * TPB) {
    const v4f* grow = (const v4f*)gumbel + (size_t)g * 4;
    v4f r0 = __builtin_nontemporal_load(grow + 0);
    v4f r1 = __builtin_nontemporal_load(grow + 1);
    v4f r2 = __builtin_nontemporal_load(grow + 2);
    v4f r3 = __builtin_nontemporal_load(grow + 3);
    v4f lg = __builtin_nontemporal_load((const v4f*)logits + g);
    v4f best = concrete_group(lg, r0, r1, r2, r3);
    __builtin_nontemporal_store(best, (v4f*)out + g);
  }
#endif
}

extern "C" void kernel_launch(void* const* d_in, const int* in_sizes, int n_in,
                              void* d_out, int out_size, void* d_ws, size_t ws_size,
                              hipStream_t stream) {
  (void)in_sizes; (void)n_in; (void)out_size; (void)d_ws; (void)ws_size;
  const float* logits = (const float*)d_in[0];  // [1024, 8192] f32
  const float* gumbel = (const float*)d_in[1];  // [1024, 2048, 4, 4] f32
  float*       out    = (float*)d_out;          // [1024, 8192] f32
  sample_concrete_kernel<<<NBLOCKS, TPB, 0, stream>>>(logits, gumbel, out);
}